// TTTLinear_29678224016185
// MI455X (gfx1250) — compile-verified
//
#include <hip/hip_runtime.h>
#include <math.h>

#define NB 4
#define NT 512
#define ND 256
#define MOMB 0.9f
#define NOVB 0.95f
#define MOM_STRIDE 260   // 256 + 4 pad: (4*row + 2*hi + {0,1}) mod 64 covers all banks

typedef float v2f __attribute__((ext_vector_type(2)));
typedef float v8f __attribute__((ext_vector_type(8)));

struct __align__(16) Smem {
  float mom[ND * MOM_STRIDE];  // momentum state, padded rows (conflict-free b64)
  float x_s[ND];               // current x_t
  float err_s[ND];             // (y - x)/D
  float yacc[2][ND];           // per-column-half partial y
  float iema[ND];              // input EMA
  float redn[8];               // novelty reduction slots (waves 0..7)
  float redf[32];              // fnorm reduction slots
  float eff_s;                 // inner_lr * lrema (broadcast scalar)
  float scale_s;               // deferred Frobenius clamp scale
};

__global__ __launch_bounds__(1024) void ttt_linear_kernel(
    const float* __restrict__ x, const float* __restrict__ Wb,
    const float* __restrict__ bias, const float* __restrict__ loglr,
    float* __restrict__ out)
{
  __shared__ Smem sm;
  const int tid  = threadIdx.x;
  const int lane = tid & 31;
  const int wid  = tid >> 5;
  const int b    = blockIdx.x;
  const int hi   = lane >> 4;            // lane half (K-pair select in A layout)
  const int rb   = (wid >> 1) << 4;      // 16-row block base
  const int row  = rb + (lane & 15);
  const int colbase = (wid & 1) << 7;    // 0 or 128

  // ---- persistent W panel in VGPRs, laid out as WMMA f32 16x16x4 A operand
  float w_u[64];
  #pragma unroll
  for (int c = 0; c < 32; ++c) {
    const int col = colbase + 4*c + 2*hi;
    const float2 wv = *reinterpret_cast<const float2*>(Wb + row*ND + col);
    w_u[2*c]   = wv.x;
    w_u[2*c+1] = wv.y;
  }
  for (int i = tid; i < ND*MOM_STRIDE; i += 1024) sm.mom[i] = 0.0f;
  if (tid < ND) sm.iema[tid] = 0.0f;

  float lrema    = 1.0f;   // thread 0 only
  float inner_lr = 0.0f;   // thread 0 only
  if (tid == 0) {
    inner_lr = fminf(fmaxf(expf(loglr[0]), 1e-5f), 1.0f);
    sm.scale_s = 1.0f;
    sm.eff_s   = 0.0f;
  }
  const float bias_r = (tid < ND) ? bias[tid] : 0.0f;
  __syncthreads();

  for (int t = 0; t < NT; ++t) {
    // ---- phase 0: load x_t, novelty partials, iema update (threads 0..255)
    if (tid < ND) {
      const float xj = x[((size_t)b*NT + t)*ND + tid];
      sm.x_s[tid] = xj;
      float a = fabsf(xj - sm.iema[tid]);          // novelty uses OLD iema
      sm.iema[tid] = NOVB*sm.iema[tid] + (1.0f-NOVB)*xj;
      #pragma unroll
      for (int off = 16; off > 0; off >>= 1) a += __shfl_xor(a, off, 32);
      if (lane == 0) sm.redn[wid] = a;
    }
    __syncthreads(); // B1

    if (tid == 0) {
      float s = 0.0f;
      #pragma unroll
      for (int i = 0; i < 8; ++i) s += sm.redn[i];
      const float nov = s * (1.0f/ND);
      const float lrs = fminf(fmaxf(1.0f + 3.0f*nov, 0.5f), 3.0f);
      lrema = NOVB*lrema + (1.0f-NOVB)*lrs;
      sm.eff_s = inner_lr * lrema;
    }

    // ---- phase 1: y_partial = W_panel * x via v_wmma_f32_16x16x4_f32
    // B[k][n] = x[k] broadcast across N -> every accumulator column == y partial
    v8f acc = {};
    #pragma unroll
    for (int c = 0; c < 32; ++c) {
      const int col = colbase + 4*c + 2*hi;
      const float2 xp = *reinterpret_cast<const float2*>(&sm.x_s[col]);
      v2f av; av.x = w_u[2*c]; av.y = w_u[2*c+1];
      v2f bv; bv.x = xp.x;     bv.y = xp.y;
      acc = __builtin_amdgcn_wmma_f32_16x16x4_f32(
          /*neg_a=*/false, av, /*neg_b=*/false, bv,
          /*c_mod=*/(short)0, acc, /*reuse_a=*/false, /*reuse_b=*/false);
    }
    if ((lane & 15) == 0) {  // lanes 0 and 16 hold rows rb+0..7 / rb+8..15
      const int mbase = rb + hi*8;
      #pragma unroll
      for (int v = 0; v < 8; ++v) sm.yacc[wid & 1][mbase + v] = acc[v];
    }
    __syncthreads(); // B2

    // ---- phase 2: y = scale*(p0+p1) + bias; err; emit output
    if (tid < ND) {
      const float yv = sm.scale_s*(sm.yacc[0][tid] + sm.yacc[1][tid]) + bias_r;
      out[((size_t)b*NT + t)*ND + tid] = yv;
      sm.err_s[tid] = (yv - sm.x_s[tid]) * (1.0f/ND);
    }
    __syncthreads(); // B3

    // ---- phase 3: momentum + W update (with deferred prev scale) + fnorm
    const float err = sm.err_s[row];
    const float eff = sm.eff_s;
    const float sp  = sm.scale_s;
    float fsum = 0.0f;
    #pragma unroll
    for (int c = 0; c < 32; ++c) {
      const int col = colbase + 4*c + 2*hi;
      const float2 xp = *reinterpret_cast<const float2*>(&sm.x_s[col]);
      float2* mptr = reinterpret_cast<float2*>(&sm.mom[row*MOM_STRIDE + col]);
      float2 mp = *mptr;
      mp.x = MOMB*mp.x + (1.0f-MOMB)*err*xp.x;
      mp.y = MOMB*mp.y + (1.0f-MOMB)*err*xp.y;
      *mptr = mp;
      const float w0 = sp*w_u[2*c]   - eff*mp.x;
      const float w1 = sp*w_u[2*c+1] - eff*mp.y;
      w_u[2*c] = w0; w_u[2*c+1] = w1;
      fsum += w0*w0 + w1*w1;
    }
    #pragma unroll
    for (int off = 16; off > 0; off >>= 1) fsum += __shfl_xor(fsum, off, 32);
    if (lane == 0) sm.redf[wid] = fsum;
    __syncthreads(); // B4

    if (tid == 0) {
      float s = 0.0f;
      #pragma unroll
      for (int i = 0; i < 32; ++i) s += sm.redf[i];
      const float fn = fmaxf(sqrtf(s), 1e-6f);
      sm.scale_s = fminf(32.0f/fn, 1.0f);   // applied lazily next step
    }
    // next iteration's B1 publishes scale_s before any consumer reads it
  }
}

extern "C" void kernel_launch(void* const* d_in, const int* in_sizes, int n_in,
                              void* d_out, int out_size, void* d_ws, size_t ws_size,
                              hipStream_t stream) {
  (void)in_sizes; (void)n_in; (void)out_size; (void)d_ws; (void)ws_size;
  const float* x    = (const float*)d_in[0];
  const float* Wb   = (const float*)d_in[1];
  const float* bias = (const float*)d_in[2];
  const float* lil  = (const float*)d_in[3];
  float* out = (float*)d_out;
  ttt_linear_kernel<<<dim3(NB), dim3(1024), 0, stream>>>(x, Wb, bias, lil, out);
}